// SmolyakIntegrator_42004780155386
// MI455X (gfx1250) — compile-verified
//
#include <hip/hip_runtime.h>

typedef __attribute__((ext_vector_type(2))) float v2f;
typedef __attribute__((ext_vector_type(8))) float v8f;

#define SMOLYAK_DIM 6

// ---------------------------------------------------------------------------
// Phase 1: accumulate quadrature moments  S[d] = sum_p w_p * x_p[d],  T = sum w_p
// Single workgroup; total point count is ~O(4000), trivially cheap.
// Writes 8 floats to ws8: [S0..S5, T, 0].
// ---------------------------------------------------------------------------
__global__ __launch_bounds__(256) void smolyak_moments_kernel(
    const int* __restrict__ ooffs,    // (nterms, 6) rule offsets (sorted-axis order)
    const int* __restrict__ olens,    // (nterms, 6) rule lengths
    const int* __restrict__ strides,  // (nterms, 6) flattening strides
    const int* __restrict__ invp,     // (nterms, 6) inverse axis permutation
    const int* __restrict__ npts,     // (nterms,)   points per term
    const float* __restrict__ nodes,
    const float* __restrict__ weights,
    int nterms,
    float* __restrict__ ws8)
{
    __shared__ float red[7][256];

    float acc[7];
#pragma unroll
    for (int c = 0; c < 7; ++c) acc[c] = 0.0f;

    for (int t = 0; t < nterms; ++t) {
        const int np = npts[t];
        const int* off = ooffs   + t * SMOLYAK_DIM;
        const int* len = olens   + t * SMOLYAK_DIM;
        const int* str = strides + t * SMOLYAK_DIM;
        const int* ivp = invp    + t * SMOLYAK_DIM;

        for (int p = (int)threadIdx.x; p < np; p += (int)blockDim.x) {
            float w = 1.0f;
            float nv[SMOLYAK_DIM];
#pragma unroll
            for (int a = 0; a < SMOLYAK_DIM; ++a) {
                const int loc = (p / str[a]) % len[a];
                const int r   = off[a] + loc;
                nv[a] = nodes[r];
                w    *= weights[r];
            }
#pragma unroll
            for (int d = 0; d < SMOLYAK_DIM; ++d)
                acc[d] += w * nv[ivp[d]];
            acc[6] += w;
        }
    }

#pragma unroll
    for (int c = 0; c < 7; ++c) red[c][threadIdx.x] = acc[c];
    __syncthreads();

    for (int s = 128; s > 0; s >>= 1) {
        if ((int)threadIdx.x < s) {
#pragma unroll
            for (int c = 0; c < 7; ++c)
                red[c][threadIdx.x] += red[c][threadIdx.x + s];
        }
        __syncthreads();
    }

    if (threadIdx.x < 8)
        ws8[threadIdx.x] = (threadIdx.x < 7) ? red[threadIdx.x][0] : 0.0f;
}

// ---------------------------------------------------------------------------
// Phase 2: out[j] = W[j,:] . S + b[j] * T   via V_WMMA_F32_16X16X4_F32.
// A[m,k] = Vext[k0+k] (replicated over rows); B[k,n] = Wext[base+n, k0+k].
// Two chained K=4 WMMAs cover the padded 8-vector [S0..S5, T, 0].
// Every D row equals the 16 tile outputs; lanes 0-15 of D vgpr0 store them.
// One wave per 16-output tile, 8 waves per block.
// ---------------------------------------------------------------------------
__global__ __launch_bounds__(256) void smolyak_gemv_wmma_kernel(
    const float* __restrict__ Wm,     // (outdim, 6) row-major
    const float* __restrict__ bias,   // (outdim,)
    const float* __restrict__ ws8,    // [S0..S5, T, 0]
    float* __restrict__ out,
    int outdim)
{
    const int lane = (int)(threadIdx.x & 31u);
    const int wave = (int)(threadIdx.x >> 5u);
    const int tile = (int)blockIdx.x * 8 + wave;
    const int base = tile * 16;
    if (base >= outdim) return;            // wave-uniform: EXEC stays all-ones for WMMA

    const int  r  = lane & 15;
    const bool hi = lane >= 16;

    const float V0 = ws8[0], V1 = ws8[1], V2 = ws8[2], V3 = ws8[3];
    const float V4 = ws8[4], V5 = ws8[5], V6 = ws8[6];

    // A fragments (16x4, f32): lanes 0-15 hold K=0,1 ; lanes 16-31 hold K=2,3.
    v2f a0, a1;
    a0.x = hi ? V2 : V0;  a0.y = hi ? V3 : V1;     // K-block 0..3
    a1.x = hi ? V6 : V4;  a1.y = hi ? 0.0f : V5;   // K-block 4..7 (T, pad)

    // B fragments (4x16, f32): lane n<16 holds K=0,1 ; lane 16+n holds K=2,3.
    const float* wrow = Wm + (size_t)(base + r) * SMOLYAK_DIM;
    v2f b0, b1;
    b0.x = hi ? wrow[2]        : wrow[0];
    b0.y = hi ? wrow[3]        : wrow[1];
    b1.x = hi ? bias[base + r] : wrow[4];          // Wext[:,6] = b
    b1.y = hi ? 0.0f           : wrow[5];          // Wext[:,7] = 0

    v8f c = {};
    c = __builtin_amdgcn_wmma_f32_16x16x4_f32(false, a0, false, b0,
                                              (short)0, c, false, false);
    c = __builtin_amdgcn_wmma_f32_16x16x4_f32(false, a1, false, b1,
                                              (short)0, c, false, false);

    if (!hi) out[base + r] = c[0];                 // D row M=0, N=lane
}

// ---------------------------------------------------------------------------
extern "C" void kernel_launch(void* const* d_in, const int* in_sizes, int n_in,
                              void* d_out, int out_size, void* d_ws, size_t ws_size,
                              hipStream_t stream)
{
    const int*   ooffs   = (const int*)d_in[0];
    const int*   olens   = (const int*)d_in[1];
    const int*   strides = (const int*)d_in[2];
    const int*   invp    = (const int*)d_in[3];
    const int*   npts    = (const int*)d_in[4];
    const float* nodes   = (const float*)d_in[5];
    const float* weights = (const float*)d_in[6];
    const float* Wm      = (const float*)d_in[7];
    const float* bias    = (const float*)d_in[8];
    float*       out     = (float*)d_out;
    float*       ws8     = (float*)d_ws;

    const int nterms = in_sizes[4];

    smolyak_moments_kernel<<<1, 256, 0, stream>>>(
        ooffs, olens, strides, invp, npts, nodes, weights, nterms, ws8);

    const int outdim = out_size;
    const int tiles  = (outdim + 15) / 16;
    const int blocks = (tiles + 7) / 8;
    smolyak_gemv_wmma_kernel<<<blocks, 256, 0, stream>>>(
        Wm, bias, ws8, out, outdim);
}